// Net_23072564314311
// MI455X (gfx1250) — compile-verified
//
#include <hip/hip_runtime.h>
#include <hip/hip_bf16.h>

typedef __attribute__((ext_vector_type(2))) float v2f;
typedef __attribute__((ext_vector_type(8))) float v8f;

#define N_NODES 50000

// ---------------- degree / normalization ----------------

__global__ void k_deg_init(float* __restrict__ deg, int n) {
    int i = blockIdx.x * blockDim.x + threadIdx.x;
    if (i < n) deg[i] = 1.0f;  // self-loop contribution
}

__global__ void k_deg_count(const int* __restrict__ col, int E, float* __restrict__ deg) {
    int e = blockIdx.x * blockDim.x + threadIdx.x;
    if (e < E) atomicAdd(&deg[col[e]], 1.0f);
}

__global__ void k_deg_rsqrt(float* __restrict__ deg, int n) {
    int i = blockIdx.x * blockDim.x + threadIdx.x;
    if (i < n) deg[i] = rsqrtf(deg[i]);  // deg >= 1 always
}

// ---------------- B pad + transpose: Bt[c*Kpad + k] = W[k*N + c], zero-padded ----------------

__global__ void k_pad_transpose(const float* __restrict__ W, float* __restrict__ Bt,
                                int K, int N, int Kpad, int Npad) {
    int idx = blockIdx.x * blockDim.x + threadIdx.x;
    if (idx >= Npad * Kpad) return;
    int c = idx / Kpad;
    int k = idx - c * Kpad;
    Bt[idx] = (c < N && k < K) ? W[(size_t)k * N + c] : 0.0f;
}

// ---------------- fp32 WMMA GEMM: C[M,N] = A[M,K] @ B[K,N] ----------------
// Bt is the zero-padded column-major copy of B: [Npad x Kpad].
// Each wave computes an MT x NT grid of 16x16 tiles (shared A/B fragments).
// Branch-free inner loop: 1 contiguous float2 load per fragment + WMMAs.
// A-layout (32-bit 16x4): lanes 0-15 -> M=lane, K={0,1}; lanes 16-31 -> M=lane-16, K={2,3}.
// C/D layout: VGPR r -> row r (lanes 0-15) / row r+8 (lanes 16-31), col = lane&15.

template <int MT, int NT>
__global__ void k_gemm_wmma_f32(const float* __restrict__ A, const float* __restrict__ Bt,
                                float* __restrict__ C, int M, int N, int K, int Kpad,
                                int n_blocks, int total_blocks) {
    const int wave = threadIdx.x >> 5;
    const int lane = threadIdx.x & 31;
    const int blk  = blockIdx.x * (blockDim.x >> 5) + wave;
    if (blk >= total_blocks) return;  // wave-uniform: EXEC stays all-1s for WMMA

    const int mb = blk / n_blocks;
    const int nb = blk % n_blocks;
    const int m0 = mb * 16 * MT;
    const int n0 = nb * 16 * NT;

    const int lm    = lane & 15;  // A row / B column / C column within a tile
    const int hi    = lane >> 4;
    const int khalf = hi * 2;     // K sub-offset {0,2}

    // Row/column base pointers, hoisted out of the K loop (row clamped: the
    // duplicated last row only feeds guarded stores).
    const float* Arow[MT];
#pragma unroll
    for (int i = 0; i < MT; ++i) {
        int r = m0 + i * 16 + lm;
        if (r > M - 1) r = M - 1;
        Arow[i] = A + (size_t)r * K;
    }
    const float* Bcol[NT];
#pragma unroll
    for (int j = 0; j < NT; ++j) {
        Bcol[j] = Bt + (size_t)(n0 + j * 16 + lm) * Kpad;  // always < Npad
    }

    v8f acc[MT][NT];
#pragma unroll
    for (int i = 0; i < MT; ++i)
#pragma unroll
        for (int j = 0; j < NT; ++j) acc[i][j] = (v8f){0, 0, 0, 0, 0, 0, 0, 0};

    const int K_main = K & ~3;
    for (int k0 = 0; k0 < K_main; k0 += 4) {
        const int k = k0 + khalf;
        v2f a[MT], b[NT];
#pragma unroll
        for (int i = 0; i < MT; ++i) {
            const float2 t = *(const float2*)(Arow[i] + k);
            a[i].x = t.x; a[i].y = t.y;
        }
#pragma unroll
        for (int j = 0; j < NT; ++j) {
            const float2 t = *(const float2*)(Bcol[j] + k);
            b[j].x = t.x; b[j].y = t.y;
        }
#pragma unroll
        for (int i = 0; i < MT; ++i)
#pragma unroll
            for (int j = 0; j < NT; ++j)
                acc[i][j] = __builtin_amdgcn_wmma_f32_16x16x4_f32(
                    false, a[i], false, b[j], (short)0, acc[i][j], false, false);
    }

    if (K_main < Kpad) {  // K % 4 != 0 remainder; Bt rows >= K are zero, so
                          // clamped (finite) A values contribute nothing.
        const int k   = K_main + khalf;
        const int kc0 = (k     < K) ? k     : K - 1;
        const int kc1 = (k + 1 < K) ? k + 1 : K - 1;
        v2f a[MT], b[NT];
#pragma unroll
        for (int i = 0; i < MT; ++i) { a[i].x = Arow[i][kc0]; a[i].y = Arow[i][kc1]; }
#pragma unroll
        for (int j = 0; j < NT; ++j) {
            const float2 t = *(const float2*)(Bcol[j] + k);
            b[j].x = t.x; b[j].y = t.y;
        }
#pragma unroll
        for (int i = 0; i < MT; ++i)
#pragma unroll
            for (int j = 0; j < NT; ++j)
                acc[i][j] = __builtin_amdgcn_wmma_f32_16x16x4_f32(
                    false, a[i], false, b[j], (short)0, acc[i][j], false, false);
    }

#pragma unroll
    for (int i = 0; i < MT; ++i)
#pragma unroll
        for (int j = 0; j < NT; ++j) {
            const int ncol = n0 + j * 16 + lm;
            if (ncol < N) {
#pragma unroll
                for (int r = 0; r < 8; ++r) {
                    const int mrow = m0 + i * 16 + r + hi * 8;
                    if (mrow < M) C[(size_t)mrow * N + ncol] = acc[i][j][r];
                }
            }
        }
}

// ---------------- aggregation ----------------

// out[i,f] = dinv[i]^2 * hlin[i,f] + bias[f]   (self-loop message + bias, fused init)
template <int F>
__global__ void k_self_bias(const float* __restrict__ hlin, const float* __restrict__ dinv,
                            const float* __restrict__ bias, float* __restrict__ out, int n) {
    int idx = blockIdx.x * blockDim.x + threadIdx.x;
    if (idx >= n * F) return;
    int i = idx / F;
    int f = idx - i * F;
    float di = dinv[i];
    out[idx] = di * di * hlin[idx] + bias[f];
}

// out[col,f] += dinv[row]*dinv[col] * hlin[row,f]
// One thread per (edge, feature): F consecutive lanes share an edge ->
// coalesced row gather and coalesced consecutive-address atomics.
template <int F>
__global__ void k_scatter(const int* __restrict__ row, const int* __restrict__ col, int E,
                          const float* __restrict__ dinv, const float* __restrict__ hlin,
                          float* __restrict__ out) {
    long long idx = (long long)blockIdx.x * blockDim.x + threadIdx.x;
    if (idx >= (long long)E * F) return;
    int e = (int)(idx / F);
    int f = (int)(idx - (long long)e * F);
    int r = row[e];
    int c = col[e];
    float nrm = dinv[r] * dinv[c];
    atomicAdd(&out[(size_t)c * F + f], nrm * hlin[(size_t)r * F + f]);
}

__global__ void k_relu(float* __restrict__ h, int n) {
    int i = blockIdx.x * blockDim.x + threadIdx.x;
    if (i < n) h[i] = fmaxf(h[i], 0.0f);
}

// ---------------- decoder: logits[e] = dot(z[src], z[dst]) over 10 dims ----------------

__global__ void k_decode(const int* __restrict__ pos, const int* __restrict__ neg, int Et,
                         const float* __restrict__ z, float* __restrict__ out, int Fout) {
    int e = blockIdx.x * blockDim.x + threadIdx.x;
    if (e >= 2 * Et) return;
    int s, d;
    if (e < Et) { s = pos[e];      d = pos[Et + e]; }
    else        { s = neg[e - Et]; d = neg[Et + (e - Et)]; }
    const float* __restrict__ zs = z + (size_t)s * Fout;
    const float* __restrict__ zd = z + (size_t)d * Fout;
    __builtin_prefetch(zd, 0, 0);  // gfx1250 global_prefetch_b8
    float acc = 0.0f;
#pragma unroll 10
    for (int f = 0; f < Fout; ++f) acc += zs[f] * zd[f];
    out[e] = acc;
}

// ---------------- driver ----------------

extern "C" void kernel_launch(void* const* d_in, const int* in_sizes, int n_in,
                              void* d_out, int out_size, void* d_ws, size_t ws_size,
                              hipStream_t stream) {
    const float* x   = (const float*)d_in[0];
    const int*   tei = (const int*)  d_in[1];  // [2, E_train]
    const int*   pei = (const int*)  d_in[2];  // [2, E_test]
    const int*   nei = (const int*)  d_in[3];  // [2, E_test]
    const float* W1  = (const float*)d_in[4];  // [512, 50]
    const float* b1  = (const float*)d_in[5];  // [50]
    const float* W2  = (const float*)d_in[6];  // [50, 10]
    const float* b2  = (const float*)d_in[7];  // [10]
    float* out = (float*)d_out;

    const int n    = N_NODES;
    const int E    = in_sizes[1] / 2;  // 1,600,000
    const int Et   = in_sizes[2] / 2;  // 100,000
    const int Fin  = in_sizes[0] / n;  // 512
    const int H    = in_sizes[5];      // 50
    const int Fout = in_sizes[7];      // 10

    const int* t_row = tei;
    const int* t_col = tei + E;

    // padded GEMM operand shapes
    const int Kp1 = (Fin + 3) & ~3;    // 512
    const int Np1 = (H + 15) & ~15;    // 64
    const int Kp2 = (H + 3) & ~3;      // 52
    const int Np2 = (Fout + 15) & ~15; // 16

    // workspace partition (floats)
    float* ws   = (float*)d_ws;
    float* dinv = ws;  ws += n;
    float* hlin = ws;  ws += (size_t)n * H;
    float* h    = ws;  ws += (size_t)n * H;
    float* zlin = ws;  ws += (size_t)n * Fout;
    float* z    = ws;  ws += (size_t)n * Fout;
    float* Bt1  = ws;  ws += (size_t)Np1 * Kp1;
    float* Bt2  = ws;

    const int TB = 256;

    // --- symmetric normalization coefficients ---
    k_deg_init <<<(n + TB - 1) / TB, TB, 0, stream>>>(dinv, n);
    k_deg_count<<<(E + TB - 1) / TB, TB, 0, stream>>>(t_col, E, dinv);
    k_deg_rsqrt<<<(n + TB - 1) / TB, TB, 0, stream>>>(dinv, n);

    // --- weight pad+transpose (tiny, once per layer) ---
    k_pad_transpose<<<(Np1 * Kp1 + TB - 1) / TB, TB, 0, stream>>>(W1, Bt1, Fin, H, Kp1, Np1);
    k_pad_transpose<<<(Np2 * Kp2 + TB - 1) / TB, TB, 0, stream>>>(W2, Bt2, H, Fout, Kp2, Np2);

    // --- layer 1: hlin = x @ W1  (fp32 WMMA, 2x2 tiles per wave) ---
    {
        const int MT = 2, NT = 2;
        int mblocks = (n + 16 * MT - 1) / (16 * MT);   // 1563
        int nblocks = Np1 / (16 * NT);                 // 2
        int total = mblocks * nblocks;                 // 3126 waves
        int wpb = 8;
        k_gemm_wmma_f32<MT, NT><<<(total + wpb - 1) / wpb, wpb * 32, 0, stream>>>(
            x, Bt1, hlin, n, H, Fin, Kp1, nblocks, total);
    }
    // --- layer 1 aggregate: h = Dsym(hlin) + b1, ReLU ---
    k_self_bias<50><<<((n * 50) + TB - 1) / TB, TB, 0, stream>>>(hlin, dinv, b1, h, n);
    {
        long long work = (long long)E * 50;
        k_scatter<50><<<(int)((work + TB - 1) / TB), TB, 0, stream>>>(t_row, t_col, E, dinv, hlin, h);
    }
    k_relu<<<((n * 50) + TB - 1) / TB, TB, 0, stream>>>(h, n * 50);

    // --- layer 2: zlin = h @ W2  (fp32 WMMA, 4x1 tiles per wave) ---
    {
        const int MT = 4, NT = 1;
        int mblocks = (n + 16 * MT - 1) / (16 * MT);   // 782
        int nblocks = Np2 / (16 * NT);                 // 1
        int total = mblocks * nblocks;
        int wpb = 8;
        k_gemm_wmma_f32<MT, NT><<<(total + wpb - 1) / wpb, wpb * 32, 0, stream>>>(
            h, Bt2, zlin, n, Fout, H, Kp2, nblocks, total);
    }
    // --- layer 2 aggregate: z = Dsym(zlin) + b2 ---
    k_self_bias<10><<<((n * 10) + TB - 1) / TB, TB, 0, stream>>>(zlin, dinv, b2, z, n);
    {
        long long work = (long long)E * 10;
        k_scatter<10><<<(int)((work + TB - 1) / TB), TB, 0, stream>>>(t_row, t_col, E, dinv, zlin, z);
    }

    // --- decoder ---
    k_decode<<<(2 * Et + TB - 1) / TB, TB, 0, stream>>>(pei, nei, Et, z, out, Fout);
}